// EdgeConv_12171937317457
// MI455X (gfx1250) — compile-verified
//
#include <hip/hip_runtime.h>

typedef __attribute__((ext_vector_type(2))) float v2f;
typedef __attribute__((ext_vector_type(8))) float v8f;

#define BATCH 8
#define NPTS  4096
#define CH    64
#define KNN   16

// ---------------- Kernel 0: squared norms per point ----------------
__global__ __launch_bounds__(256)
void norms_kernel(const float* __restrict__ x, float* __restrict__ norms) {
  int p = blockIdx.x * blockDim.x + threadIdx.x;  // 0..B*N-1
  if (p >= BATCH * NPTS) return;
  const float* xr = x + (size_t)p * CH;
  float s = 0.f;
#pragma unroll
  for (int c = 0; c < CH; c += 4) {
    float4 v = *(const float4*)(xr + c);
    s += v.x * v.x + v.y * v.y + v.z * v.z + v.w * v.w;
  }
  norms[p] = s;
}

// ---------------- Kernel 1: fused Gram (fp32 WMMA) + top-16 ----------------
// Block = 128 threads (4 waves); 16 rows of one batch per block; columns swept
// in 64-wide chunks (one 16x16 fp32-WMMA dot tile per wave). B-operand
// registers are double-buffered so next chunk's global loads overlap this
// chunk's WMMAs AND the barrier/selection phase. Selection is 4x parallel:
// 64 threads keep per-(row,quarter) partial top-16 lists; merged once at end.
// Ordering key: cn[col] - 2*dot (row norm drops out of the per-row top-k).
__global__ __launch_bounds__(128)
void knn_kernel(const float* __restrict__ x, const float* __restrict__ norms,
                int* __restrict__ knn) {
  const int b       = blockIdx.y;
  const int rowbase = blockIdx.x * 16;
  const int tid     = threadIdx.x;
  const int lane    = tid & 31;
  const int wave    = tid >> 5;      // 0..3
  const int mrow    = lane & 15;     // M (row) / N (col) index within tile
  const int khalf   = lane >> 4;     // K half select per f32 WMMA layout

  __shared__ float dtile[16][68];    // 68-f32 stride: padded + 16B aligned
  __shared__ float cn[64];
  __shared__ float mdist[3][16][KNN];  // partial lists of quarters 1..3
  __shared__ int   midx [3][16][KNN];

  const float* xbat = x + (size_t)b * NPTS * CH;

  // A operand: 16 rows x 64 features resident in registers for the whole sweep.
  // f32 16x4 A layout: lane m=l%16 holds K = kc*4 + (l/16)*2 + {0,1}.
  const float* xr = xbat + (size_t)(rowbase + mrow) * CH;
  v2f a_reg[16];
#pragma unroll
  for (int kc = 0; kc < 16; ++kc)
    a_reg[kc] = *(const v2f*)(xr + kc * 4 + khalf * 2);

  // Register-resident sorted top-16 (ascending). Used by tid<64.
  float bd[KNN]; int bi[KNN];
#pragma unroll
  for (int i = 0; i < KNN; ++i) { bd[i] = 3.4e38f; bi[i] = 0; }

  auto insert = [&](float cand, int ci) {
    if (cand < bd[KNN - 1]) {
      bd[KNN - 1] = cand; bi[KNN - 1] = ci;
#pragma unroll
      for (int q = KNN - 1; q >= 1; --q) {   // one bubble pass re-sorts
        if (bd[q] < bd[q - 1]) {
          float td = bd[q]; bd[q] = bd[q - 1]; bd[q - 1] = td;
          int   ti = bi[q]; bi[q] = bi[q - 1]; bi[q - 1] = ti;
        }
      }
    }
  };

  const int selrow = tid & 15;   // row owned during selection
  const int selq   = tid >> 4;   // column quarter owned (tid<64 -> 0..3)

  // One chunk: WMMA on `cur` B-regs, prefetch chunk `cbn` into `nxt`.
  auto do_chunk = [&](int cb, v2f (&cur)[16], v2f (&nxt)[16], int cbn) {
    if (tid < 64) cn[tid] = norms[b * NPTS + cb + tid];

    if (cbn < NPTS) {                       // uniform scalar branch
      const float* xn = xbat + (size_t)(cbn + wave * 16 + mrow) * CH;
#pragma unroll
      for (int kc = 0; kc < 16; ++kc)
        nxt[kc] = *(const v2f*)(xn + kc * 4 + khalf * 2);
    }

    v8f acc = {};
#pragma unroll
    for (int kc = 0; kc < 16; ++kc)
      acc = __builtin_amdgcn_wmma_f32_16x16x4_f32(false, a_reg[kc], false,
                                                  cur[kc], (short)0, acc,
                                                  false, false);
    // C/D layout: VGPR i -> row (i + khalf*8), col = lane%16.
#pragma unroll
    for (int i = 0; i < 8; ++i)
      dtile[i + khalf * 8][wave * 16 + mrow] = -2.0f * acc[i];

    __syncthreads();

    if (tid < 64) {
      const float* drow = dtile[selrow] + selq * 16;
      const float* crow = cn + selq * 16;
      const int    base = cb + selq * 16;
#pragma unroll 1
      for (int j4 = 0; j4 < 16; j4 += 4) {   // unroll 1: keep code small
        float4 dv = *(const float4*)(drow + j4);
        float4 cv = *(const float4*)(crow + j4);
        insert(cv.x + dv.x, base + j4 + 0);
        insert(cv.y + dv.y, base + j4 + 1);
        insert(cv.z + dv.z, base + j4 + 2);
        insert(cv.w + dv.w, base + j4 + 3);
      }
    }
    __syncthreads();
  };

  // Prime buffer A with chunk 0, then pipeline in pairs (no buffer-swap movs).
  v2f bufA[16], bufB[16];
  {
    const float* xc0 = xbat + (size_t)(wave * 16 + mrow) * CH;
#pragma unroll
    for (int kc = 0; kc < 16; ++kc)
      bufA[kc] = *(const v2f*)(xc0 + kc * 4 + khalf * 2);
  }
#pragma unroll 1
  for (int cb = 0; cb < NPTS; cb += 128) {
    do_chunk(cb,      bufA, bufB, cb + 64);
    do_chunk(cb + 64, bufB, bufA, cb + 128);
  }

  // Merge: quarters 1..3 publish lists; row thread folds them into quarter 0.
  if (tid >= 16 && tid < 64) {
#pragma unroll
    for (int i = 0; i < KNN; ++i) {
      mdist[selq - 1][selrow][i] = bd[i];
      midx [selq - 1][selrow][i] = bi[i];
    }
  }
  __syncthreads();

  if (tid < 16) {
#pragma unroll 1
    for (int q = 0; q < 3; ++q) {
#pragma unroll 1
      for (int i = 0; i < KNN; ++i)
        insert(mdist[q][tid][i], midx[q][tid][i]);
    }
    int* o = knn + ((size_t)(b * NPTS + rowbase + tid)) * KNN;
#pragma unroll
    for (int i = 0; i < KNN; ++i) o[i] = bi[i];
  }
}

// ---------------- Kernel 2: gather + feat(16x128) @ W(128x64), ReLU, max_K ---
// Block = 256 threads (8 waves); W staged in LDS; one wave per point.
// All xj/xi global loads are hoisted into one register-resident prefetch so
// the 128-WMMA K-loop is fed only from registers + LDS.
__global__ __launch_bounds__(256)
void edgeconv_kernel(const float* __restrict__ x, const float* __restrict__ W,
                     const float* __restrict__ bias, const int* __restrict__ knn,
                     float* __restrict__ out) {
  __shared__ float Wl[128 * 64];
  const int tid = threadIdx.x;
  for (int i = tid; i < 128 * 64; i += 256) Wl[i] = W[i];
  __syncthreads();

  const int lane  = tid & 31;
  const int wave  = tid >> 5;                 // 0..7
  const int p     = blockIdx.x * 8 + wave;    // point id (exact: 4096*8 = B*N)
  const int b     = p >> 12;                  // / NPTS
  const int mrow  = lane & 15;                // neighbor index j
  const int khalf = lane >> 4;

  const int nbr = knn[(size_t)p * KNN + mrow];
  const float* xi = x + (size_t)p * CH;
  const float* xj = x + ((size_t)(b * NPTS + nbr)) * CH;

  // Prefetch all A-operand data (one load clause, one wait).
  v2f vjr[16], vir[16];
#pragma unroll
  for (int kc = 0; kc < 16; ++kc) {
    const int k0 = kc * 4 + khalf * 2;
    vjr[kc] = *(const v2f*)(xj + k0);
    vir[kc] = *(const v2f*)(xi + k0);
  }

  v8f acc[4] = {v8f{}, v8f{}, v8f{}, v8f{}};

  // First 64 K-dims: feat = xj - xi
#pragma unroll
  for (int kc = 0; kc < 16; ++kc) {
    const int k0 = kc * 4 + khalf * 2;
    v2f av = vjr[kc] - vir[kc];
    const float* wr = Wl + (size_t)k0 * 64 + mrow;
#pragma unroll
    for (int nt = 0; nt < 4; ++nt) {
      v2f bv = { wr[nt * 16], wr[64 + nt * 16] };
      acc[nt] = __builtin_amdgcn_wmma_f32_16x16x4_f32(false, av, false, bv,
                                                      (short)0, acc[nt],
                                                      false, false);
    }
  }
  // Second 64 K-dims: feat = xi (same K offsets - 64, cached in vir)
#pragma unroll
  for (int kc = 16; kc < 32; ++kc) {
    const int k0 = kc * 4 + khalf * 2;   // 64..127
    v2f av = vir[kc - 16];
    const float* wr = Wl + (size_t)k0 * 64 + mrow;
#pragma unroll
    for (int nt = 0; nt < 4; ++nt) {
      v2f bv = { wr[nt * 16], wr[64 + nt * 16] };
      acc[nt] = __builtin_amdgcn_wmma_f32_16x16x4_f32(false, av, false, bv,
                                                      (short)0, acc[nt],
                                                      false, false);
    }
  }

  // max over 16 neighbors; bias + ReLU commute with max (monotone).
#pragma unroll
  for (int nt = 0; nt < 4; ++nt) {
    float m = acc[nt][0];
#pragma unroll
    for (int i = 1; i < 8; ++i) m = fmaxf(m, acc[nt][i]);
    m = fmaxf(m, __shfl_xor(m, 16, 32));          // rows 0-7 vs 8-15
    m = fmaxf(m + bias[nt * 16 + mrow], 0.0f);
    if (lane < 16) out[(size_t)p * CH + nt * 16 + lane] = m;
  }
}

// ---------------- launcher ----------------
extern "C" void kernel_launch(void* const* d_in, const int* in_sizes, int n_in,
                              void* d_out, int out_size, void* d_ws, size_t ws_size,
                              hipStream_t stream) {
  (void)in_sizes; (void)n_in; (void)out_size; (void)ws_size;
  const float* x    = (const float*)d_in[0];
  const float* W    = (const float*)d_in[1];
  const float* bias = (const float*)d_in[2];
  float* out = (float*)d_out;

  // workspace: [norms: B*N f32][knn: B*N*16 i32]  (~2.2 MB)
  float* norms = (float*)d_ws;
  int*   knn   = (int*)((char*)d_ws + (size_t)BATCH * NPTS * sizeof(float));

  norms_kernel<<<(BATCH * NPTS + 255) / 256, 256, 0, stream>>>(x, norms);
  dim3 g1(NPTS / 16, BATCH);
  knn_kernel<<<g1, 128, 0, stream>>>(x, norms, knn);
  edgeconv_kernel<<<(BATCH * NPTS) / 8, 256, 0, stream>>>(x, W, bias, knn, out);
}